// ScriptableAC_2954937500154
// MI455X (gfx1250) — compile-verified
//
#include <hip/hip_runtime.h>

// B=65536, F=512, H=256, A=4, P=2, E=32.
// No activations => every 2-layer MLP collapses to one affine map.
// Fused problem: C[B,16] = features[B,512] @ Wall[512,16]  (cols: t*5+j,
// j=0 value, j=1..4 logits; col 15 = zero pad), bias folded into accumulator
// init, task-2 pointgoal correction via tiny Gp[2,5] in the epilogue.

typedef __attribute__((ext_vector_type(2))) float v2f;
typedef __attribute__((ext_vector_type(8))) float v8f;

// workspace layout (float offsets)
#define WS_WALL 0      // 8192 floats: fused [512,16] weights
#define WS_BIAS 8192   // 16 floats : fused bias per column
#define WS_GP   8208   // 10 floats : Gp[p*5+j] pointgoal coeffs (task 2)
#define WS_WE2A 8224   // 128 floats: a2_W1[512:544] @ a2_W2  -> [32,4]
#define WS_WE2C 8352   // 32 floats : c2_W1[512:544] @ c2_W2  -> [32,1]

__global__ __launch_bounds__(256) void collapse_wall(
    const float* __restrict__ a0_W1, const float* __restrict__ a0_W2,
    const float* __restrict__ a1_W1, const float* __restrict__ a1_W2,
    const float* __restrict__ a2_W1, const float* __restrict__ a2_W2,
    const float* __restrict__ c0_W1, const float* __restrict__ c0_W2,
    const float* __restrict__ c1_W1, const float* __restrict__ c1_W2,
    const float* __restrict__ c2_W1, const float* __restrict__ c2_W2,
    float* __restrict__ Wall) {
  int idx = blockIdx.x * 256 + threadIdx.x;  // 0..8191  -> (f, col)
  int f = idx >> 4, col = idx & 15;
  float s = 0.f;
  if (col < 15) {
    int t = col / 5, j = col % 5;
    const float *W1, *W2;
    int ld, jc;
    if (j == 0) {  // critic column
      W1 = (t == 0) ? c0_W1 : (t == 1) ? c1_W1 : c2_W1;
      W2 = (t == 0) ? c0_W2 : (t == 1) ? c1_W2 : c2_W2;
      ld = 1; jc = 0;
    } else {       // actor logit column
      W1 = (t == 0) ? a0_W1 : (t == 1) ? a1_W1 : a2_W1;
      W2 = (t == 0) ? a0_W2 : (t == 1) ? a1_W2 : a2_W2;
      ld = 4; jc = j - 1;
    }
    // task-2 W1 is [544,256]; rows 0..511 are the features part (concat order)
    const float* r = W1 + (size_t)f * 256;
    for (int h = 0; h < 256; ++h) s += r[h] * W2[h * ld + jc];
  }
  Wall[idx] = s;
}

__global__ __launch_bounds__(256) void collapse_small(
    const float* __restrict__ a2_Wp, const float* __restrict__ a2_bp,
    const float* __restrict__ a2_W1, const float* __restrict__ a2_b1,
    const float* __restrict__ a2_W2, const float* __restrict__ a2_b2,
    const float* __restrict__ c2_Wp, const float* __restrict__ c2_bp,
    const float* __restrict__ c2_W1, const float* __restrict__ c2_b1,
    const float* __restrict__ c2_W2, const float* __restrict__ c2_b2,
    const float* __restrict__ a0_b1, const float* __restrict__ a0_W2,
    const float* __restrict__ a0_b2, const float* __restrict__ a1_b1,
    const float* __restrict__ a1_W2, const float* __restrict__ a1_b2,
    const float* __restrict__ c0_b1, const float* __restrict__ c0_W2,
    const float* __restrict__ c0_b2, const float* __restrict__ c1_b1,
    const float* __restrict__ c1_W2, const float* __restrict__ c1_b2,
    float* __restrict__ ws) {
  int tid = threadIdx.x;
  float* We2a = ws + WS_WE2A;
  float* We2c = ws + WS_WE2C;
  float* Gp   = ws + WS_GP;
  float* bias = ws + WS_BIAS;

  // Phase A: embed-rows of W1 composed with W2
  if (tid < 128) {
    int e = tid >> 2, jj = tid & 3;
    const float* r = a2_W1 + (size_t)(512 + e) * 256;
    float s = 0.f;
    for (int h = 0; h < 256; ++h) s += r[h] * a2_W2[h * 4 + jj];
    We2a[tid] = s;
  } else if (tid < 160) {
    int e = tid - 128;
    const float* r = c2_W1 + (size_t)(512 + e) * 256;
    float s = 0.f;
    for (int h = 0; h < 256; ++h) s += r[h] * c2_W2[h];
    We2c[e] = s;
  }
  __syncthreads();

  // Phase B1: pointgoal coefficients Gp[p*5+j]
  if (tid < 10) {
    int p = tid / 5, j = tid % 5;
    float s = 0.f;
    if (j == 0) { for (int e = 0; e < 32; ++e) s += c2_Wp[p * 32 + e] * We2c[e]; }
    else        { for (int e = 0; e < 32; ++e) s += a2_Wp[p * 32 + e] * We2a[e * 4 + (j - 1)]; }
    Gp[tid] = s;
  }
  // Phase B2: fused biases per column
  if (tid >= 16 && tid < 32) {
    int col = tid - 16;
    float s = 0.f;
    if (col < 15) {
      int t = col / 5, j = col % 5;
      if (j == 0) {
        const float* b1 = (t == 0) ? c0_b1 : (t == 1) ? c1_b1 : c2_b1;
        const float* W2 = (t == 0) ? c0_W2 : (t == 1) ? c1_W2 : c2_W2;
        const float* b2 = (t == 0) ? c0_b2 : (t == 1) ? c1_b2 : c2_b2;
        s = b2[0];
        for (int h = 0; h < 256; ++h) s += b1[h] * W2[h];
        if (t == 2) for (int e = 0; e < 32; ++e) s += c2_bp[e] * We2c[e];
      } else {
        int jc = j - 1;
        const float* b1 = (t == 0) ? a0_b1 : (t == 1) ? a1_b1 : a2_b1;
        const float* W2 = (t == 0) ? a0_W2 : (t == 1) ? a1_W2 : a2_W2;
        const float* b2 = (t == 0) ? a0_b2 : (t == 1) ? a1_b2 : a2_b2;
        s = b2[jc];
        for (int h = 0; h < 256; ++h) s += b1[h] * W2[h * 4 + jc];
        if (t == 2) for (int e = 0; e < 32; ++e) s += a2_bp[e] * We2a[e * 4 + jc];
      }
    }
    bias[col] = s;
  }
}

// Main GEMM: 8 waves/block, one 16x16 output tile per wave (16 rows x 16 cols),
// K=512 via 128x V_WMMA_F32_16X16X4_F32. B matrix staged in LDS, interleaved
// in K-pairs so a B fragment is a single 8-byte LDS read.
__global__ __launch_bounds__(256) void ac_gemm(
    const float* __restrict__ features, const float* __restrict__ pointgoal,
    const int* __restrict__ task_id, const float* __restrict__ ws,
    float* __restrict__ out) {
  __shared__ float lds[8192];
  const float* Wall = ws + WS_WALL;
  const float* bias = ws + WS_BIAS;
  const float* Gp   = ws + WS_GP;
  int tid = threadIdx.x;

  // stage Wall[k][n] -> lds[(k>>1)*32 + n*2 + (k&1)]
  const float4* W4 = (const float4*)Wall;
#pragma unroll
  for (int i = 0; i < 8; ++i) {
    int i4 = tid + i * 256;
    float4 v = W4[i4];
    float vv[4] = {v.x, v.y, v.z, v.w};
    int base = i4 * 4;
#pragma unroll
    for (int c = 0; c < 4; ++c) {
      int id = base + c;
      int k = id >> 4, n = id & 15;
      lds[((k >> 1) << 5) + (n << 1) + (k & 1)] = vv[c];
    }
  }
  __syncthreads();

  int wave = tid >> 5, lane = tid & 31;
  int n = lane & 15;        // A: row M ; B/C: column N
  int half = lane >> 4;     // K sub-offset selector (0 -> K+0/1, 1 -> K+2/3)
  size_t rowA = (size_t)blockIdx.x * 128 + wave * 16 + n;
  const float* arow = features + rowA * 512 + half * 2;
  int bbase = half * 32 + n * 2;

  v8f acc;
  float bv = bias[n];       // fold bias into accumulator (per column)
#pragma unroll
  for (int i = 0; i < 8; ++i) acc[i] = bv;

#pragma unroll 4
  for (int k0 = 0; k0 < 512; k0 += 4) {
    if ((k0 & 31) == 0) __builtin_prefetch(arow + k0 + 256, 0, 0);
    v2f a = *(const v2f*)(arow + k0);             // A[M=n, k0+half*2 .. +1]
    v2f b = *(const v2f*)(lds + k0 * 16 + bbase); // B[k0+half*2 .. +1, N=n]
    acc = __builtin_amdgcn_wmma_f32_16x16x4_f32(
        false, a, false, b, (short)0, acc, false, false);
  }

  __syncthreads();  // done reading Wall from LDS everywhere; reuse for tiles
#pragma unroll
  for (int i = 0; i < 8; ++i) {
    int m = half * 8 + i;                       // C layout: VGPR i -> M = i (+8)
    lds[(wave * 16 + m) * 16 + n] = acc[i];
  }
  __syncthreads();

  // epilogue: per-row task select + task-2 pointgoal correction
  if (tid < 128) {
    size_t row = (size_t)blockIdx.x * 128 + tid;
    int t = task_id[row];
    int cb = t * 5;
    float p0 = 0.f, p1 = 0.f;
    if (t == 2) { p0 = pointgoal[row * 2]; p1 = pointgoal[row * 2 + 1]; }
    float* o = out + row * 5;
#pragma unroll
    for (int j = 0; j < 5; ++j) {
      float v = lds[tid * 16 + cb + j];
      if (t == 2) v += p0 * Gp[j] + p1 * Gp[5 + j];
      o[j] = v;
    }
  }
}

extern "C" void kernel_launch(void* const* d_in, const int* in_sizes, int n_in,
                              void* d_out, int out_size, void* d_ws, size_t ws_size,
                              hipStream_t stream) {
  (void)in_sizes; (void)n_in; (void)out_size; (void)ws_size;
  const float* features = (const float*)d_in[0];
  const float* pointgoal = (const float*)d_in[1];
  const float* a0_W1 = (const float*)d_in[2];
  const float* a0_b1 = (const float*)d_in[3];
  const float* a0_W2 = (const float*)d_in[4];
  const float* a0_b2 = (const float*)d_in[5];
  const float* a1_W1 = (const float*)d_in[6];
  const float* a1_b1 = (const float*)d_in[7];
  const float* a1_W2 = (const float*)d_in[8];
  const float* a1_b2 = (const float*)d_in[9];
  const float* a2_Wp = (const float*)d_in[10];
  const float* a2_bp = (const float*)d_in[11];
  const float* a2_W1 = (const float*)d_in[12];
  const float* a2_b1 = (const float*)d_in[13];
  const float* a2_W2 = (const float*)d_in[14];
  const float* a2_b2 = (const float*)d_in[15];
  const float* c0_W1 = (const float*)d_in[16];
  const float* c0_b1 = (const float*)d_in[17];
  const float* c0_W2 = (const float*)d_in[18];
  const float* c0_b2 = (const float*)d_in[19];
  const float* c1_W1 = (const float*)d_in[20];
  const float* c1_b1 = (const float*)d_in[21];
  const float* c1_W2 = (const float*)d_in[22];
  const float* c1_b2 = (const float*)d_in[23];
  const float* c2_Wp = (const float*)d_in[24];
  const float* c2_bp = (const float*)d_in[25];
  const float* c2_W1 = (const float*)d_in[26];
  const float* c2_b1 = (const float*)d_in[27];
  const float* c2_W2 = (const float*)d_in[28];
  const float* c2_b2 = (const float*)d_in[29];
  const int* task_id = (const int*)d_in[30];
  float* ws = (float*)d_ws;
  float* out = (float*)d_out;

  collapse_wall<<<32, 256, 0, stream>>>(a0_W1, a0_W2, a1_W1, a1_W2, a2_W1, a2_W2,
                                        c0_W1, c0_W2, c1_W1, c1_W2, c2_W1, c2_W2,
                                        ws + WS_WALL);
  collapse_small<<<1, 256, 0, stream>>>(a2_Wp, a2_bp, a2_W1, a2_b1, a2_W2, a2_b2,
                                        c2_Wp, c2_bp, c2_W1, c2_b1, c2_W2, c2_b2,
                                        a0_b1, a0_W2, a0_b2, a1_b1, a1_W2, a1_b2,
                                        c0_b1, c0_W2, c0_b2, c1_b1, c1_W2, c1_b2,
                                        ws);
  ac_gemm<<<512, 256, 0, stream>>>(features, pointgoal, task_id, ws, out);
}